// SparseCrop_42279658062072
// MI455X (gfx1250) — compile-verified
//
#include <hip/hip_runtime.h>

typedef __attribute__((ext_vector_type(4))) float v4f;
typedef __attribute__((ext_vector_type(4))) int   v4i;

#ifndef CROP_MIN
#define CROP_MIN 10
#define CROP_MAX 90
#endif

// Streaming mask-multiply kernel.
// Each thread handles one float4 (16 bytes) of the (N,64) feature matrix per
// loop iteration: 16 lanes = one row, so a wave32 covers two rows per access.
// feats/out are read/written once -> non-temporal hints keep the 2x256MB
// streams from thrashing the 192MB L2. coords (16MB, reused by 16 lanes) use
// default (RT) caching. A one-stride-ahead global_prefetch_b8 on the feats
// stream hides HBM latency in the grid-stride loop.
__global__ __launch_bounds__(256) void SparseCrop_42279658062072_kernel(
    const int* __restrict__ coords,    // (N,4) int32
    const float* __restrict__ feats,   // (N,64) f32
    float* __restrict__ out_feats,     // (N,64) f32
    float* __restrict__ out_mask,      // (N,)   f32 (0.0 / 1.0)
    long total4)                       // N * 16 float4-chunks
{
    const long stride = (long)gridDim.x * (long)blockDim.x;
    for (long g = (long)blockIdx.x * blockDim.x + threadIdx.x; g < total4; g += stride) {
        // Prefetch the feats chunk this thread will consume next iteration
        // (emits global_prefetch_b8 on gfx1250; speculative, dropped if OOB,
        //  but guard anyway to keep the address stream sane).
        if (g + stride < total4) {
            __builtin_prefetch((const char*)feats + (g + stride) * 16, 0, 0);
        }

        const long row = g >> 4;         // feature row (0..N-1)
        const int  c4  = (int)(g & 15);  // which float4 within the 64-wide row

        // Aligned 16B coord load: coords row is {x,y,z,batch}, 16B-aligned.
        v4i c = *((const v4i*)coords + row);
        const int m = (c.x >= CROP_MIN) & (c.x < CROP_MAX) &
                      (c.y >= CROP_MIN) & (c.y < CROP_MAX) &
                      (c.z >= CROP_MIN) & (c.z < CROP_MAX);
        const float s = m ? 1.0f : 0.0f;

        // Read-once stream: NT load (th:TH_LOAD_NT), b128.
        v4f f = __builtin_nontemporal_load((const v4f*)feats + g);
        v4f o = f * s;
        // Write-once stream: NT store (th:TH_STORE_NT), b128.
        __builtin_nontemporal_store(o, (v4f*)out_feats + g);

        // One lane per row emits the mask value.
        if (c4 == 0) {
            __builtin_nontemporal_store(s, out_mask + row);
        }
    }
}

extern "C" void kernel_launch(void* const* d_in, const int* in_sizes, int n_in,
                              void* d_out, int out_size, void* d_ws, size_t ws_size,
                              hipStream_t stream) {
    (void)n_in; (void)d_ws; (void)ws_size; (void)out_size;

    const int*   coords = (const int*)d_in[0];    // (N,4) int32
    const float* feats  = (const float*)d_in[1];  // (N,64) f32

    const long N = (long)in_sizes[0] / 4;         // coords flat count / 4
    const long C = 64;                            // reference constant

    float* out_feats = (float*)d_out;             // first N*64 floats
    float* out_mask  = (float*)d_out + N * C;     // then N mask floats

    const long total4 = N * (C / 4);              // float4 chunks = N*16

    const int  block = 256;                       // 8 wave32s per block
    long want = (total4 + block - 1) / block;
    int  grid = (int)(want < 8192 ? (want > 0 ? want : 1) : 8192);

    SparseCrop_42279658062072_kernel<<<grid, block, 0, stream>>>(
        coords, feats, out_feats, out_mask, total4);
}